// ScaledDotProductAttention_37168646979887
// MI455X (gfx1250) — compile-verified
//
#include <hip/hip_runtime.h>
#include <hip/hip_bf16.h>

// Scaled dot-product attention, fp32, returning (out, attn).
// B=16, Lq=Lk=2048, D=128 (fixed by the harness/reference).
//
// One workgroup = one (batch, 16-row Q tile). The full 16x2048 score strip is
// staged in LDS (CDNA5 WGP has 320KB LDS; we use ~137KB), softmax is done
// in-LDS, attn is written once, and out = P*V is computed from LDS P with
// fp32 WMMA (V_WMMA_F32_16X16X4_F32) so numerics match the fp32 reference.

typedef __attribute__((ext_vector_type(2))) float v2f;
typedef __attribute__((ext_vector_type(8))) float v8f;

#define ATT_B   16
#define ATT_LQ  2048
#define ATT_LK  2048
#define ATT_D   128
#define QTILE   16
#define NWAVES  4
#define BLOCK   (NWAVES * 32)

__global__ __launch_bounds__(BLOCK)
void attn_fused_wmma_kernel(const float* __restrict__ q,
                            const float* __restrict__ k,
                            const float* __restrict__ v,
                            const int*   __restrict__ mask,
                            float* __restrict__ out,
                            float* __restrict__ attn)
{
    extern __shared__ float smem[];
    float* sQ   = smem;                         // QTILE x D       (2048 f)
    float* sS   = smem + QTILE * ATT_D;         // QTILE x LK      (32768 f)
    float* sRed = sS + QTILE * ATT_LK;          // 256 f reduction scratch

    const int b    = blockIdx.y;
    const int q0   = blockIdx.x * QTILE;
    const int tid  = threadIdx.x;
    const int wave = tid >> 5;
    const int lane = tid & 31;
    const int laneN  = lane & 15;   // A-frag: row M.  B-frag / C-frag: col N.
    const int khalf  = lane >> 4;   // which K half of the 16x4 / 4x16 fragment

    // ---------------- load Q tile into LDS (contiguous 16x128 block) -------
    const float* qtile = q + ((size_t)b * ATT_LQ + q0) * ATT_D;
    for (int i = tid; i < QTILE * ATT_D; i += BLOCK) sQ[i] = qtile[i];
    __syncthreads();

    const float inv_temper = 0.08838834764831845f; // 1/sqrt(128)
    const float* kb = k + (size_t)b * ATT_LK * ATT_D;

    // ---------------- S = Q K^T * inv_temper + (-1e9)*(1-mask) -------------
    // 128 N-tiles of 16 keys; waves stride across them.
    for (int nt = wave; nt < ATT_LK / 16; nt += NWAVES) {
        const int n0 = nt * 16;
        if (n0 + NWAVES * 16 < ATT_LK)  // prefetch next K tile for this wave
            __builtin_prefetch(kb + (size_t)(n0 + NWAVES * 16 + laneN) * ATT_D, 0, 0);

        v8f acc = {};
        const float* arow = sQ + laneN * ATT_D + khalf * 2;                  // Q[M][kd..kd+1]
        const float* krow = kb + (size_t)(n0 + laneN) * ATT_D + khalf * 2;   // K[n][kd..kd+1]
        #pragma unroll 4
        for (int kk = 0; kk < ATT_D / 4; ++kk) {
            v2f afrag = { arow[0], arow[1] };
            v2f bfrag = { krow[0], krow[1] };
            acc = __builtin_amdgcn_wmma_f32_16x16x4_f32(
                false, afrag, false, bfrag, (short)0, acc, false, false);
            arow += 4;
            krow += 4;
        }
        // C layout: vgpr i holds row M = i + 8*khalf, col N = laneN.
        #pragma unroll
        for (int i = 0; i < 8; ++i) {
            const int m   = i + khalf * 8;
            const int col = n0 + laneN;
            float s = acc[i] * inv_temper;
            const int mk = mask[((size_t)b * ATT_LQ + q0 + m) * ATT_LK + col];
            s += -1.0e9f * (1.0f - (float)mk);
            sS[m * ATT_LK + col] = s;
        }
    }
    __syncthreads();

    // ---------------- softmax over LK, 8 threads per row -------------------
    const int row = tid >> 3;   // 0..15
    const int sub = tid & 7;    // 0..7
    float mx = -3.4e38f;
    for (int c = sub; c < ATT_LK; c += 8) mx = fmaxf(mx, sS[row * ATT_LK + c]);
    sRed[tid] = mx;
    __syncthreads();
    if (sub == 0) {
        float m2 = sRed[tid];
        #pragma unroll
        for (int j = 1; j < 8; ++j) m2 = fmaxf(m2, sRed[tid + j]);
        sRed[tid] = m2;
    }
    __syncthreads();
    const float rmax = sRed[row << 3];

    float sum = 0.f;
    for (int c = sub; c < ATT_LK; c += 8) {
        const float e = __expf(sS[row * ATT_LK + c] - rmax);
        sS[row * ATT_LK + c] = e;
        sum += e;
    }
    sRed[128 + tid] = sum;
    __syncthreads();
    if (sub == 0) {
        float s2 = 0.f;
        #pragma unroll
        for (int j = 0; j < 8; ++j) s2 += sRed[128 + tid + j];
        sRed[128 + tid] = s2;
    }
    __syncthreads();
    const float rinv = 1.0f / sRed[128 + (row << 3)];

    float* attn_row = attn + ((size_t)b * ATT_LQ + q0 + row) * ATT_LK;
    for (int c = sub; c < ATT_LK; c += 8) {
        const float p = sS[row * ATT_LK + c] * rinv;
        sS[row * ATT_LK + c] = p;     // keep normalized P in LDS for P*V
        attn_row[c] = p;              // stream attn to HBM once
    }
    __syncthreads();

    // ---------------- out = P (16 x LK) * V (LK x D) -----------------------
    const float* vb = v + (size_t)b * ATT_LK * ATT_D;
    for (int dt = wave; dt < ATT_D / 16; dt += NWAVES) {
        const int d0 = dt * 16;
        v8f acc = {};
        const float* prow = sS + laneN * ATT_LK + khalf * 2;                 // P[M][kr..kr+1]
        const float* vcol = vb + (size_t)(khalf * 2) * ATT_D + d0 + laneN;   // V[kr][N], V[kr+1][N]
        #pragma unroll 4
        for (int kk = 0; kk < ATT_LK / 4; ++kk) {
            v2f afrag = { prow[0], prow[1] };
            v2f bfrag = { vcol[0], vcol[ATT_D] };
            acc = __builtin_amdgcn_wmma_f32_16x16x4_f32(
                false, afrag, false, bfrag, (short)0, acc, false, false);
            prow += 4;
            vcol += 4 * ATT_D;
        }
        #pragma unroll
        for (int i = 0; i < 8; ++i) {
            const int m = i + khalf * 8;
            out[((size_t)b * ATT_LQ + q0 + m) * ATT_D + d0 + laneN] = acc[i];
        }
    }
}

extern "C" void kernel_launch(void* const* d_in, const int* in_sizes, int n_in,
                              void* d_out, int out_size, void* d_ws, size_t ws_size,
                              hipStream_t stream)
{
    (void)in_sizes; (void)n_in; (void)out_size; (void)d_ws; (void)ws_size;

    const float* q    = (const float*)d_in[0];
    const float* k    = (const float*)d_in[1];
    const float* v    = (const float*)d_in[2];
    const int*   mask = (const int*)d_in[3];

    float* out  = (float*)d_out;                                  // B*Lq*D
    float* attn = out + (size_t)ATT_B * ATT_LQ * ATT_D;           // B*Lq*Lk

    dim3 grid(ATT_LQ / QTILE, ATT_B);
    dim3 block(BLOCK);
    const size_t shmem =
        (size_t)(QTILE * ATT_D + QTILE * ATT_LK + 256) * sizeof(float);

    attn_fused_wmma_kernel<<<grid, block, shmem, stream>>>(q, k, v, mask, out, attn);
}